// GNNs_31052613550443
// MI455X (gfx1250) — compile-verified
//
#include <hip/hip_runtime.h>
#include <hip/hip_bf16.h>

#define NND   50000
#define NE    600000
#define NG    1024
#define HD    128

typedef __attribute__((ext_vector_type(2))) float v2f;
typedef __attribute__((ext_vector_type(8))) float v8f;

// ---------------- utility ----------------
__global__ void k_zero(float* __restrict__ p, int n) {
  int i = blockIdx.x * blockDim.x + threadIdx.x;
  if (i < n) p[i] = 0.f;
}

// ---------------- degrees ----------------
__global__ void k_edge_deg(const int* __restrict__ src, const int* __restrict__ dst,
                           float* __restrict__ dout, float* __restrict__ din, int E) {
  int e = blockIdx.x * blockDim.x + threadIdx.x;
  if (e >= E) return;
  atomicAdd(dout + src[e], 1.f);
  atomicAdd(din  + dst[e], 1.f);
}

__global__ void k_rsqrt_deg(float* __restrict__ p, int n) {
  int i = blockIdx.x * blockDim.x + threadIdx.x;
  if (i < n) p[i] = rsqrtf(fmaxf(p[i], 1.f));
}

// ---------------- embedding lookup ----------------
__global__ void k_embed(const int* __restrict__ nf, const float* __restrict__ emb,
                        float* __restrict__ h0, int total) {
  int t = blockIdx.x * blockDim.x + threadIdx.x;
  if (t >= total) return;
  int n = t >> 7, c = t & (HD - 1);
  h0[t] = emb[nf[n] * HD + c];
}

// ---------------- edge scatter: agg[dst] += h[src] * oi[src] ----------------
// one thread per (edge, 4 channels): 128-bit loads, f32 global atomics (L2-resident agg)
__global__ void k_scatter(const float* __restrict__ h, const float* __restrict__ oi,
                          const int* __restrict__ src, const int* __restrict__ dst,
                          float* __restrict__ agg, int E) {
  unsigned t = blockIdx.x * blockDim.x + threadIdx.x;
  unsigned e = t >> 5;
  if (e >= (unsigned)E) return;
  int c4 = (t & 31u) << 2;
  int s = src[e], d = dst[e];
  float sc = oi[s];
  const float4 v = *(const float4*)(h + (size_t)s * HD + c4);
  float* ap = agg + (size_t)d * HD + c4;
  atomicAdd(ap + 0, v.x * sc);
  atomicAdd(ap + 1, v.y * sc);
  atomicAdd(ap + 2, v.z * sc);
  atomicAdd(ap + 3, v.w * sc);
}

// ---------------- WMMA f32 GEMM: Out = act((A .* rowscale) @ W + bias) ----------------
// one wave32 per 16x16 tile; K-loop of V_WMMA_F32_16X16X4_F32
__global__ void k_wmma_gemm(const float* __restrict__ A, const float* __restrict__ W,
                            const float* __restrict__ bias, const float* __restrict__ rowscale,
                            float* __restrict__ Out, int M, int K, int N,
                            int apply_act, float slope) {
  int tile_m = blockIdx.x;
  int tile_n = blockIdx.y;
  int lane = threadIdx.x;        // 0..31
  int half = lane >> 4;          // 0: K+{0,1}, 1: K+{2,3}  (ISA 16x4 f32 A layout)
  int l16  = lane & 15;
  int row  = tile_m * 16 + l16;  // A row this lane streams (M = lane%16)
  int col  = tile_n * 16 + l16;  // B/D column this lane owns (N = lane%16)
  float rs = rowscale ? rowscale[row] : 1.0f;
  const float* Arow = A + (size_t)row * K;
  v8f c = {};
  for (int k0 = 0; k0 < K; k0 += 4) {
    int ka = k0 + (half << 1);   // half 0 -> K0,K0+1 ; half 1 -> K0+2,K0+3
    v2f a, b;
    a.x = Arow[ka]     * rs;
    a.y = Arow[ka + 1] * rs;
    b.x = W[(size_t)(ka)     * N + col];
    b.y = W[(size_t)(ka + 1) * N + col];
    c = __builtin_amdgcn_wmma_f32_16x16x4_f32(false, a, false, b, (short)0, c,
                                              false, false);
  }
  float bc = bias ? bias[col] : 0.f;
  // D layout: VGPR v, lane -> M = v + 8*half, N = lane%16
#pragma unroll
  for (int v = 0; v < 8; ++v) {
    int m = tile_m * 16 + v + (half << 3);
    float val = c[v] + bc;
    if (apply_act) val = (val >= 0.f) ? val : slope * val;
    Out[(size_t)m * N + tile_n * 16 + l16] = val;
  }
}

// ---------------- graph mean pooling ----------------
__global__ void k_pool(const float* __restrict__ h, const int* __restrict__ gid,
                       float* __restrict__ hg, float* __restrict__ cnt, int total) {
  int t = blockIdx.x * blockDim.x + threadIdx.x;
  if (t >= total) return;
  int n = t >> 7, c = t & (HD - 1);
  int g = gid[n];
  atomicAdd(hg + (size_t)g * HD + c, h[t]);
  if (c == 0) atomicAdd(cnt + g, 1.f);
}

__global__ void k_pool_div(float* __restrict__ hg, const float* __restrict__ cnt, int total) {
  int t = blockIdx.x * blockDim.x + threadIdx.x;
  if (t >= total) return;
  hg[t] /= fmaxf(cnt[t >> 7], 1.f);
}

// ---------------- batchnorm (training-mode batch stats) ----------------
__global__ void k_bn_stats(const float* __restrict__ x, float* __restrict__ mu,
                           float* __restrict__ rsig, int rows, int C) {
  int c = blockIdx.x;
  float s = 0.f, s2 = 0.f;
  for (int r = threadIdx.x; r < rows; r += blockDim.x) {
    float v = x[(size_t)r * C + c];
    s += v; s2 += v * v;
  }
  __shared__ float sh[256], sh2[256];
  sh[threadIdx.x] = s; sh2[threadIdx.x] = s2;
  __syncthreads();
  for (int off = 128; off > 0; off >>= 1) {
    if ((int)threadIdx.x < off) {
      sh[threadIdx.x]  += sh[threadIdx.x + off];
      sh2[threadIdx.x] += sh2[threadIdx.x + off];
    }
    __syncthreads();
  }
  if (threadIdx.x == 0) {
    float m = sh[0] / rows;
    float var = sh2[0] / rows - m * m;
    mu[c] = m;
    rsig[c] = rsqrtf(var + 1e-5f);
  }
}

__global__ void k_bn_apply(const float* __restrict__ x, const float* __restrict__ mu,
                           const float* __restrict__ rsig, const float* __restrict__ g,
                           const float* __restrict__ b, float* __restrict__ y,
                           int total, int C, float slope) {
  int t = blockIdx.x * blockDim.x + threadIdx.x;
  if (t >= total) return;
  int c = t & (C - 1);
  float v = (x[t] - mu[c]) * rsig[c] * g[c] + b[c];
  y[t] = (v >= 0.f) ? v : slope * v;
}

// ---------------- final 32->2 matmul + log_softmax ----------------
__global__ void k_head(const float* __restrict__ h4, const float* __restrict__ w,
                       const float* __restrict__ b, float* __restrict__ out, int G) {
  int gidx = blockIdx.x * blockDim.x + threadIdx.x;
  if (gidx >= G) return;
  const float* hr = h4 + (size_t)gidx * 32;
  float y0 = b[0], y1 = b[1];
#pragma unroll
  for (int k = 0; k < 32; ++k) {
    float v = hr[k];
    y0 += v * w[k * 2 + 0];
    y1 += v * w[k * 2 + 1];
  }
  float m = fmaxf(y0, y1);
  float lse = m + logf(expf(y0 - m) + expf(y1 - m));
  out[gidx * 2 + 0] = y0 - lse;
  out[gidx * 2 + 1] = y1 - lse;
}

// ---------------- host-side orchestration ----------------
static inline int cdiv(long long a, int b) { return (int)((a + b - 1) / b); }

extern "C" void kernel_launch(void* const* d_in, const int* in_sizes, int n_in,
                              void* d_out, int out_size, void* d_ws, size_t ws_size,
                              hipStream_t stream) {
  const int*   nf    = (const int*)d_in[0];
  const int*   esrc  = (const int*)d_in[1];
  const int*   edst  = (const int*)d_in[2];
  const int*   gid   = (const int*)d_in[3];
  const float* emb   = (const float*)d_in[4];
  const float* W1    = (const float*)d_in[5];
  const float* b1    = (const float*)d_in[6];
  const float* W2    = (const float*)d_in[7];
  const float* b2    = (const float*)d_in[8];
  const float* e1w   = (const float*)d_in[9];
  const float* e1b   = (const float*)d_in[10];
  const float* e2w   = (const float*)d_in[11];
  const float* e2b   = (const float*)d_in[12];
  const float* e3w   = (const float*)d_in[13];
  const float* e3b   = (const float*)d_in[14];
  const float* e4w   = (const float*)d_in[15];
  const float* e4b   = (const float*)d_in[16];
  const float* e5w   = (const float*)d_in[17];
  const float* e5b   = (const float*)d_in[18];
  const float* bn0g  = (const float*)d_in[19];
  const float* bn0b  = (const float*)d_in[20];
  const float* bn1g  = (const float*)d_in[21];
  const float* bn1b  = (const float*)d_in[22];
  float* out = (float*)d_out;

  float* w = (float*)d_ws;
  size_t o = 0;
  float* oi  = w + o; o += NND;
  float* ii  = w + o; o += NND;
  float* h0  = w + o; o += (size_t)NND * HD;
  float* agg = w + o; o += (size_t)NND * HD;
  float* hb  = w + o; o += (size_t)NND * HD;   // h1 then h2
  float* hg  = w + o; o += (size_t)NG * HD;
  float* cnt = w + o; o += NG;
  float* x1  = w + o; o += (size_t)NG * HD;
  float* h1g = w + o; o += (size_t)NG * HD;
  float* x2  = w + o; o += (size_t)NG * HD;
  float* h2g = w + o; o += (size_t)NG * HD;
  float* h3  = w + o; o += (size_t)NG * 64;
  float* h4  = w + o; o += (size_t)NG * 32;
  float* mu  = w + o; o += HD;
  float* rsg = w + o; o += HD;

  const int B = 256;
  const long long NH = (long long)NND * HD;

  // degrees -> D_out^{-1/2}, D_in^{-1/2}
  k_zero<<<cdiv(2 * NND, B), B, 0, stream>>>(oi, 2 * NND);   // oi,ii contiguous
  k_edge_deg<<<cdiv(NE, B), B, 0, stream>>>(esrc, edst, oi, ii, NE);
  k_rsqrt_deg<<<cdiv(2 * NND, B), B, 0, stream>>>(oi, 2 * NND);

  // h0 = emb[node_feat]
  k_embed<<<cdiv(NH, B), B, 0, stream>>>(nf, emb, h0, (int)NH);

  // layer 1: agg = scatter(h0 * oi); h1 = relu((agg * ii) @ W1 + b1)
  k_zero<<<cdiv(NH, B), B, 0, stream>>>(agg, (int)NH);
  k_scatter<<<cdiv((long long)NE * 32, B), B, 0, stream>>>(h0, oi, esrc, edst, agg, NE);
  k_wmma_gemm<<<dim3(NND / 16, HD / 16), 32, 0, stream>>>(agg, W1, b1, ii, hb,
                                                          NND, HD, HD, 1, 0.0f);
  // layer 2
  k_zero<<<cdiv(NH, B), B, 0, stream>>>(agg, (int)NH);
  k_scatter<<<cdiv((long long)NE * 32, B), B, 0, stream>>>(hb, oi, esrc, edst, agg, NE);
  k_wmma_gemm<<<dim3(NND / 16, HD / 16), 32, 0, stream>>>(agg, W2, b2, ii, hb,
                                                          NND, HD, HD, 1, 0.0f);

  // mean pooling per graph
  k_zero<<<cdiv(NG * HD + NG, B), B, 0, stream>>>(hg, NG * HD + NG);  // hg,cnt contiguous
  k_pool<<<cdiv(NH, B), B, 0, stream>>>(hb, gid, hg, cnt, (int)NH);
  k_pool_div<<<cdiv(NG * HD, B), B, 0, stream>>>(hg, cnt, NG * HD);

  // MLP head with batchnorm + leaky relu
  k_wmma_gemm<<<dim3(NG / 16, HD / 16), 32, 0, stream>>>(hg, e1w, e1b, nullptr, x1,
                                                         NG, HD, HD, 0, 0.0f);
  k_bn_stats<<<HD, 256, 0, stream>>>(x1, mu, rsg, NG, HD);
  k_bn_apply<<<cdiv(NG * HD, B), B, 0, stream>>>(x1, mu, rsg, bn0g, bn0b, h1g,
                                                 NG * HD, HD, 0.05f);
  k_wmma_gemm<<<dim3(NG / 16, HD / 16), 32, 0, stream>>>(h1g, e2w, e2b, nullptr, x2,
                                                         NG, HD, HD, 0, 0.0f);
  k_bn_stats<<<HD, 256, 0, stream>>>(x2, mu, rsg, NG, HD);
  k_bn_apply<<<cdiv(NG * HD, B), B, 0, stream>>>(x2, mu, rsg, bn1g, bn1b, h2g,
                                                 NG * HD, HD, 0.05f);
  k_wmma_gemm<<<dim3(NG / 16, 64 / 16), 32, 0, stream>>>(h2g, e3w, e3b, nullptr, h3,
                                                         NG, HD, 64, 1, 0.1f);
  k_wmma_gemm<<<dim3(NG / 16, 32 / 16), 32, 0, stream>>>(h3, e4w, e4b, nullptr, h4,
                                                         NG, 64, 32, 1, 0.1f);
  k_head<<<cdiv(NG, B), B, 0, stream>>>(h4, e5w, e5b, out, NG);
}